// DMM_45878840656347
// MI455X (gfx1250) — compile-verified
//
#include <hip/hip_runtime.h>

typedef __attribute__((ext_vector_type(2))) float v2f;
typedef __attribute__((ext_vector_type(4))) float v4f;
typedef __attribute__((ext_vector_type(8))) float v8f;

#define DIMSZ 100
#define CTXN  8
#define K1N   6
#define NW1   200001   /* N_WORDS + 1 : rows of W, cols of WP */
#define HPAD  104      /* LDS row stride (multiple of 8 dwords for b64/b128 align) */

// Extract c[idx] for idx in [0,8) without dynamic vector indexing (cndmask tree).
__device__ __forceinline__ float diag_extract(const v8f& c, int idx) {
    float v = c[0];
#pragma unroll
    for (int j = 1; j < 8; ++j)
        if (idx == j) v = c[j];
    return v;
}

__global__ __launch_bounds__(64) void cbow_wmma_score_kernel(
    const int*   __restrict__ ctxs,     // [B, 8]
    const int*   __restrict__ docs,     // [B]
    const int*   __restrict__ targets,  // [B, 6]
    const float* __restrict__ D,        // [N_DOCS, 100]
    const float* __restrict__ W,        // [200001, 100]
    const float* __restrict__ WP,       // [100, 200001]
    float*       __restrict__ out,      // [B, 6]
    int B)
{
    __shared__ float hsm[16 * HPAD];

    const int tid  = threadIdx.x;   // 0..63
    const int wave = tid >> 5;      // 0..1
    const int lane = tid & 31;
    const int b0   = blockIdx.x * 16;

    // ---------------- Phase 1: h tile [16 x 100] (f32) into LDS ----------------
    // wave w builds rows i = 8w .. 8w+7 ; lanes 0..24 each own one float4 chunk.
    for (int i = wave * 8; i < wave * 8 + 8; ++i) {
        const int b   = min(b0 + i, B - 1);
        const int doc = docs[b];                 // uniform -> s_load
        int widx[CTXN];
#pragma unroll
        for (int c = 0; c < CTXN; ++c)
            widx[c] = ctxs[b * CTXN + c];        // uniform -> s_load

        if (lane < 25) {                         // 25 * float4 == 100 floats
            const v4f* __restrict__ drow =
                (const v4f*)(D + (long long)doc * DIMSZ);
            v4f acc = __builtin_nontemporal_load(drow + lane);  // D: streamed, no reuse
#pragma unroll
            for (int c = 0; c < CTXN; ++c) {
                const v4f* __restrict__ wrow =
                    (const v4f*)(W + (long long)widx[c] * DIMSZ);
                acc += wrow[lane];
            }
            *(v4f*)(&hsm[i * HPAD + lane * 4]) = acc;
        }
    }
    __syncthreads();

    // ---------------- Phase 2: scores via V_WMMA_F32_16X16X4_F32 ----------------
    // Tile matmul per k: Dmat[m,n] = h[b0+m] . WP[:, targets[b0+n, k]],
    // accumulate over 25 chunks of K=4; result wanted on the diagonal m==n.
    const int n    = lane & 15;      // matrix column == b-row within tile
    const int half = lane >> 4;      // 0 -> K rows {0,1}, 1 -> K rows {2,3}
    const int bn   = min(b0 + n, B - 1);

    int tcol[3];
#pragma unroll
    for (int kk = 0; kk < 3; ++kk)
        tcol[kk] = targets[bn * K1N + (wave * 3 + kk)];

    v8f C0 = {}, C1 = {}, C2 = {};

    for (int kb = 0; kb < 25; ++kb) {
        const int r0 = kb * 4 + half * 2;        // this lane's K-row pair
        // A fragment (16x4 f32): VGPR0 = K0|K2, VGPR1 = K1|K3, M = lane%16
        v2f a;
        a.x = hsm[n * HPAD + r0];
        a.y = hsm[n * HPAD + r0 + 1];
        // B fragments (4x16 f32, mirrored layout): rows r0, r0+1 of WP at column tcol
        const float* p0 = WP + (long long)r0 * NW1 + tcol[0];
        const float* p1 = WP + (long long)r0 * NW1 + tcol[1];
        const float* p2 = WP + (long long)r0 * NW1 + tcol[2];
        v2f bf0, bf1, bf2;
        bf0.x = p0[0]; bf0.y = p0[NW1];
        bf1.x = p1[0]; bf1.y = p1[NW1];
        bf2.x = p2[0]; bf2.y = p2[NW1];

        C0 = __builtin_amdgcn_wmma_f32_16x16x4_f32(false, a, false, bf0,
                                                   (short)0, C0, false, false);
        C1 = __builtin_amdgcn_wmma_f32_16x16x4_f32(false, a, false, bf1,
                                                   (short)0, C1, false, false);
        C2 = __builtin_amdgcn_wmma_f32_16x16x4_f32(false, a, false, bf2,
                                                   (short)0, C2, false, false);
    }

    // Diagonal extraction:
    //  lanes 0..15 hold D[m=v, n=lane]      -> diag valid for lane < 8 at v = lane
    //  lanes 16..31 hold D[m=v+8, n=lane-16]-> diag valid for lane >= 24 at v = lane-24
    int vidx = -1;
    if (lane < 8)        vidx = lane;
    else if (lane >= 24) vidx = lane - 24;

    if (vidx >= 0 && (b0 + n) < B) {
        float* o = out + (long long)(b0 + n) * K1N + wave * 3;
        o[0] = diag_extract(C0, vidx);
        o[1] = diag_extract(C1, vidx);
        o[2] = diag_extract(C2, vidx);
    }
}

extern "C" void kernel_launch(void* const* d_in, const int* in_sizes, int n_in,
                              void* d_out, int out_size, void* d_ws, size_t ws_size,
                              hipStream_t stream) {
    const int*   ctxs    = (const int*)d_in[0];
    const int*   docs    = (const int*)d_in[1];
    const int*   targets = (const int*)d_in[2];
    const float* D       = (const float*)d_in[3];
    const float* W       = (const float*)d_in[4];
    const float* WP      = (const float*)d_in[5];
    float*       out     = (float*)d_out;

    const int B     = in_sizes[1];          // docs is [B]
    const int tiles = (B + 15) / 16;        // 1024 for B=16384

    hipLaunchKernelGGL(cbow_wmma_score_kernel, dim3(tiles), dim3(64), 0, stream,
                       ctxs, docs, targets, D, W, WP, out, B);
}